// GlobalLinearSelfAttention_12223476924599
// MI455X (gfx1250) — compile-verified
//
#include <hip/hip_runtime.h>
#include <math.h>

typedef __attribute__((ext_vector_type(2))) float v2f;
typedef __attribute__((ext_vector_type(4))) float v4f;
typedef __attribute__((ext_vector_type(8))) float v8f;

#define TM 64
#define TN 64
#define TK 16
#define LDAS 20   // padded LDS stride for A tile [64][16]
#define LDBS 68   // padded LDS stride for B/K tiles [16][64]

__device__ __forceinline__ v8f wmma_f32(v2f a, v2f b, v8f c) {
  // V_WMMA_F32_16X16X4_F32: (neg_a, A, neg_b, B, c_mod, C, reuse_a, reuse_b)
  return __builtin_amdgcn_wmma_f32_16x16x4_f32(false, a, false, b, (short)0, c,
                                               false, false);
}

// ---------------------------------------------------------------------------
// Generic NN GEMM tile: C[64x64] = A[64xK] * B[Kx64], all row-major.
// 256 threads = 8 waves; wave w -> m0 = (w&3)*16, n0 = (w>>2)*32 (two 16x16 accs)
// ---------------------------------------------------------------------------
__device__ __forceinline__ void gemm_nn_tile(const float* __restrict__ A, int lda,
                                             const float* __restrict__ B, int ldb,
                                             float* __restrict__ C, int ldc,
                                             int K, int mBase, int nBase) {
  __shared__ float As[TM * LDAS];
  __shared__ float Bs[TK * LDBS];

  const int tid  = threadIdx.x;
  const int lane = tid & 31;
  const int wave = tid >> 5;
  const int lm   = lane & 15;
  const int half = lane >> 4;            // 0: K={0,1}, 1: K={2,3}
  const int m0w  = (wave & 3) * 16;
  const int n0w  = (wave >> 2) * 32;

  const int am = tid >> 2;               // 0..63  (A tile row)
  const int ak = (tid & 3) * 4;          // 0..12  (A tile k group)
  const int bk = tid >> 4;               // 0..15  (B tile k row)
  const int bn = (tid & 15) * 4;         // 0..60  (B tile n group)

  v8f acc0 = {};
  v8f acc1 = {};

  for (int kb = 0; kb < K; kb += TK) {
    v4f av = *(const v4f*)(A + (size_t)(mBase + am) * lda + kb + ak);
    v4f bv = *(const v4f*)(B + (size_t)(kb + bk) * ldb + nBase + bn);
    __syncthreads();
    *(v4f*)(As + am * LDAS + ak) = av;
    *(v4f*)(Bs + bk * LDBS + bn) = bv;
    __syncthreads();
#pragma unroll
    for (int kk = 0; kk < TK; kk += 4) {
      const int kr = kk + 2 * half;
      v2f a, b0, b1;
      a.x  = As[(m0w + lm) * LDAS + kr];
      a.y  = As[(m0w + lm) * LDAS + kr + 1];
      b0.x = Bs[kr * LDBS + n0w + lm];
      b0.y = Bs[(kr + 1) * LDBS + n0w + lm];
      b1.x = Bs[kr * LDBS + n0w + 16 + lm];
      b1.y = Bs[(kr + 1) * LDBS + n0w + 16 + lm];
      acc0 = wmma_f32(a, b0, acc0);
      acc1 = wmma_f32(a, b1, acc1);
    }
  }
#pragma unroll
  for (int i = 0; i < 8; ++i) {
    const int row = mBase + m0w + i + half * 8;   // lanes16-31 hold rows i+8
    const int col = nBase + n0w + lm;
    C[(size_t)row * ldc + col]      = acc0[i];
    C[(size_t)row * ldc + col + 16] = acc1[i];
  }
}

// ---------------------------------------------------------------------------
// TN GEMM tile: C[64x64] = A^T * B, with A stored [K x 64] (lda), B [K x 64].
// Used for context = k_norm^T @ v  (split-K).
// ---------------------------------------------------------------------------
__device__ __forceinline__ void gemm_tn_tile(const float* __restrict__ A, int lda,
                                             const float* __restrict__ B, int ldb,
                                             float* __restrict__ C, int ldc, int K) {
  __shared__ float As[TK * LDBS];
  __shared__ float Bs[TK * LDBS];

  const int tid  = threadIdx.x;
  const int lane = tid & 31;
  const int wave = tid >> 5;
  const int lm   = lane & 15;
  const int half = lane >> 4;
  const int m0w  = (wave & 3) * 16;
  const int n0w  = (wave >> 2) * 32;

  const int rk = tid >> 4;          // 0..15
  const int rn = (tid & 15) * 4;    // 0..60

  v8f acc0 = {};
  v8f acc1 = {};

  for (int kb = 0; kb < K; kb += TK) {
    v4f av = *(const v4f*)(A + (size_t)(kb + rk) * lda + rn);
    v4f bv = *(const v4f*)(B + (size_t)(kb + rk) * ldb + rn);
    __syncthreads();
    *(v4f*)(As + rk * LDBS + rn) = av;
    *(v4f*)(Bs + rk * LDBS + rn) = bv;
    __syncthreads();
#pragma unroll
    for (int kk = 0; kk < TK; kk += 4) {
      const int kr = kk + 2 * half;
      v2f a, b0, b1;
      a.x  = As[kr * LDBS + m0w + lm];        // A^T: row m from column of As
      a.y  = As[(kr + 1) * LDBS + m0w + lm];
      b0.x = Bs[kr * LDBS + n0w + lm];
      b0.y = Bs[(kr + 1) * LDBS + n0w + lm];
      b1.x = Bs[kr * LDBS + n0w + 16 + lm];
      b1.y = Bs[(kr + 1) * LDBS + n0w + 16 + lm];
      acc0 = wmma_f32(a, b0, acc0);
      acc1 = wmma_f32(a, b1, acc1);
    }
  }
#pragma unroll
  for (int i = 0; i < 8; ++i) {
    const int row = m0w + i + half * 8;
    const int col = n0w + lm;
    C[(size_t)row * ldc + col]      = acc0[i];
    C[(size_t)row * ldc + col + 16] = acc1[i];
  }
}

// ---------------------------------------------------------------------------
__global__ void gemm_nn_kernel(const float* __restrict__ A, int lda,
                               const float* __restrict__ B, int ldb,
                               float* __restrict__ C, int ldc, int K) {
  gemm_nn_tile(A, lda, B, ldb, C, ldc, K, blockIdx.y * TM, blockIdx.x * TN);
}

// q softmax over d=64 (within head), * 64^-0.5. One wave per row.
__global__ void q_softmax_kernel(float* __restrict__ qkv) {
  const int tid  = threadIdx.x;
  const int lane = tid & 31;
  const int wave = tid >> 5;
  const size_t r  = (size_t)blockIdx.x * 8 + wave;   // row id in [0, 2*4096*16)
  const size_t bn = r >> 4;
  const int    h  = (int)(r & 15);
  float* p = qkv + bn * 3072 + h * 64;

  float x0 = p[lane], x1 = p[lane + 32];
  float m = fmaxf(x0, x1);
#pragma unroll
  for (int off = 16; off; off >>= 1) m = fmaxf(m, __shfl_xor(m, off, 32));
  float e0 = __expf(x0 - m), e1 = __expf(x1 - m);
  float s = e0 + e1;
#pragma unroll
  for (int off = 16; off; off >>= 1) s += __shfl_xor(s, off, 32);
  const float inv = 0.125f / s;   // fold scale = d^-0.5 = 1/8
  p[lane]      = e0 * inv;
  p[lane + 32] = e1 * inv;
}

// ---------------------------------------------------------------------------
// k softmax over n=4096 per column d, parallelized:
//   k_stats:  grid (32 heads * 16 slices); each block does rows [slice*256, +256)
//             and produces per-slice online-softmax partials (max, sumexp).
//   k_norm:   same grid; combines the 16 partials (fixed order -> deterministic)
//             and normalizes its slice in place.
// ---------------------------------------------------------------------------
__global__ void k_stats_kernel(const float* __restrict__ qkv,
                               float* __restrict__ pmax,
                               float* __restrict__ psum) {
  __shared__ float lm[4 * 64];
  __shared__ float ls[4 * 64];
  const int bx = blockIdx.x;             // bh*16 + slice
  const int bh = bx >> 4, slice = bx & 15;
  const int b = bh >> 4, h = bh & 15;
  const int tid = threadIdx.x;
  const int d = tid & 63, grp = tid >> 6;          // 4 row-groups of 64 rows
  const float* base = qkv + (size_t)b * 4096 * 3072 + 1024 + h * 64 + d;
  const int n0 = slice * 256 + grp * 64;

  float m = -3.402823466e38f;
  for (int r = 0; r < 64; ++r) m = fmaxf(m, base[(size_t)(n0 + r) * 3072]);
  float s = 0.f;
  for (int r = 0; r < 64; ++r) s += __expf(base[(size_t)(n0 + r) * 3072] - m);
  lm[grp * 64 + d] = m;
  ls[grp * 64 + d] = s;
  __syncthreads();
  if (tid < 64) {
    float M = lm[tid], S = ls[tid];
#pragma unroll
    for (int g = 1; g < 4; ++g) {
      float m2 = lm[g * 64 + tid], s2 = ls[g * 64 + tid];
      float nm = fmaxf(M, m2);
      S = S * __expf(M - nm) + s2 * __expf(m2 - nm);
      M = nm;
    }
    pmax[(size_t)bx * 64 + tid] = M;
    psum[(size_t)bx * 64 + tid] = S;
  }
}

__global__ void k_norm_kernel(float* __restrict__ qkv,
                              const float* __restrict__ pmax,
                              const float* __restrict__ psum) {
  __shared__ float gm[64];
  __shared__ float gi[64];
  const int bx = blockIdx.x;
  const int bh = bx >> 4, slice = bx & 15;
  const int b = bh >> 4, h = bh & 15;
  const int tid = threadIdx.x;
  const int d = tid & 63, grp = tid >> 6;

  if (tid < 64) {
    float M = pmax[(size_t)(bh * 16) * 64 + tid];
    float S = psum[(size_t)(bh * 16) * 64 + tid];
    for (int sl = 1; sl < 16; ++sl) {           // fixed order: deterministic
      float m2 = pmax[(size_t)(bh * 16 + sl) * 64 + tid];
      float s2 = psum[(size_t)(bh * 16 + sl) * 64 + tid];
      float nm = fmaxf(M, m2);
      S = S * __expf(M - nm) + s2 * __expf(m2 - nm);
      M = nm;
    }
    gm[tid] = M;
    gi[tid] = 1.0f / S;
  }
  __syncthreads();
  const float M = gm[d], invS = gi[d];
  float* base = qkv + (size_t)b * 4096 * 3072 + 1024 + h * 64 + d;
  const int n0 = slice * 256 + grp * 64;
  for (int r = 0; r < 64; ++r) {
    const size_t idx = (size_t)(n0 + r) * 3072;
    base[idx] = __expf(base[idx] - M) * invS;
  }
}

// ---------------------------------------------------------------------------
// context = k_norm^T @ v per head, split-K by 8 (grid 32 x 8 = 256 blocks)
// ---------------------------------------------------------------------------
__global__ void head_context_kernel(const float* __restrict__ qkv,
                                    float* __restrict__ ctxp) {
  const int bh = blockIdx.x, sp = blockIdx.y;
  const int b = bh >> 4, h = bh & 15;
  const size_t rowOff = (size_t)sp * 512 * 3072;
  const float* Kp = qkv + (size_t)b * 4096 * 3072 + 1024 + h * 64 + rowOff;
  const float* Vp = qkv + (size_t)b * 4096 * 3072 + 2048 + h * 64 + rowOff;
  float* C = ctxp + ((size_t)bh * 8 + sp) * 4096;
  gemm_tn_tile(Kp, 3072, Vp, 3072, C, 64, 512);
}

__global__ void ctx_combine_kernel(const float* __restrict__ ctxp,
                                   float* __restrict__ ctx) {
  const int bh = blockIdx.x;
  const int tid = threadIdx.x;
#pragma unroll
  for (int j = 0; j < 16; ++j) {
    const int e = j * 256 + tid;
    float s = 0.f;
#pragma unroll
    for (int sp = 0; sp < 8; ++sp) s += ctxp[((size_t)bh * 8 + sp) * 4096 + e];
    ctx[(size_t)bh * 4096 + e] = s;
  }
}

// attn[b, n, h*64+e] = q_sm[b,n,h,:] @ context[bh]
__global__ void head_out_kernel(const float* __restrict__ qkv,
                                const float* __restrict__ ctx,
                                float* __restrict__ attn) {
  const int bh = blockIdx.y, b = bh >> 4, h = bh & 15;
  const float* A = qkv + (size_t)b * 4096 * 3072 + h * 64;   // q region
  const float* B = ctx + (size_t)bh * 4096;
  float* C = attn + (size_t)b * 4096 * 1024 + h * 64;
  gemm_nn_tile(A, 3072, B, 64, C, 1024, 64, blockIdx.x * TM, 0);
}

// ---------------------------------------------------------------------------
extern "C" void kernel_launch(void* const* d_in, const int* in_sizes, int n_in,
                              void* d_out, int out_size, void* d_ws, size_t ws_size,
                              hipStream_t stream) {
  (void)in_sizes; (void)n_in; (void)out_size; (void)ws_size;
  const float* feats = (const float*)d_in[0];   // [2,4096,1024]
  const float* W_qkv = (const float*)d_in[1];   // [1024,3072]
  const float* W_out = (const float*)d_in[2];   // [1024,1024]
  float* out = (float*)d_out;                   // [2,4096,1024]

  float* qkv  = (float*)d_ws;                          // 8192*3072   (100.7 MB)
  float* kmax = qkv + (size_t)8192 * 3072;             // 32*16*64
  float* ksum = kmax + (size_t)32 * 16 * 64;           // 32*16*64
  float* ctxp = ksum + (size_t)32 * 16 * 64;           // 32*8*64*64  (4 MB)
  float* ctx  = ctxp + (size_t)32 * 8 * 4096;          // 32*64*64
  float* attn = ctx + (size_t)32 * 4096;               // 8192*1024   (33.6 MB)

  const dim3 blk(256);

  // 1) qkv = feats @ W_qkv   (M=8192, N=3072, K=1024)
  gemm_nn_kernel<<<dim3(3072 / TN, 8192 / TM), blk, 0, stream>>>(
      feats, 1024, W_qkv, 3072, qkv, 3072, 1024);

  // 2) q <- softmax_d(q) * d^-0.5   (131072 rows, 8 rows/block)
  q_softmax_kernel<<<131072 / 8, 256, 0, stream>>>(qkv);

  // 3) k <- softmax_n(k): parallel two-pass online softmax (512 blocks each)
  k_stats_kernel<<<32 * 16, 256, 0, stream>>>(qkv, kmax, ksum);
  k_norm_kernel<<<32 * 16, 256, 0, stream>>>(qkv, kmax, ksum);

  // 4) context = k^T @ v per head, split-K=8 (256 blocks) + combine
  head_context_kernel<<<dim3(32, 8), blk, 0, stream>>>(qkv, ctxp);
  ctx_combine_kernel<<<32, 256, 0, stream>>>(ctxp, ctx);

  // 5) attn = q @ context per head
  head_out_kernel<<<dim3(4096 / TM, 32), blk, 0, stream>>>(qkv, ctx, attn);

  // 6) out = attn @ W_out   (M=8192, N=1024, K=1024)
  gemm_nn_kernel<<<dim3(1024 / TN, 8192 / TM), blk, 0, stream>>>(
      attn, 1024, W_out, 1024, out, 1024, 1024);
}